// pyg_Hetero_GCNLayer_2010044694737
// MI455X (gfx1250) — compile-verified
//
#include <hip/hip_runtime.h>
#include <hip/hip_bf16.h>

typedef __attribute__((ext_vector_type(2))) float v2f;
typedef __attribute__((ext_vector_type(8))) float v8f;

#define Dfeat 128

// One wave (32 lanes) computes one 16x16 tile of Out = X @ Wm^T + bias [opt relu].
// blockDim = 256 (8 waves); wave w handles N-tile w (H = 128 = 8 * 16).
// gridDim.x = nrows / 16 (nrows is a multiple of 16).
__global__ void gemm_bias_wmma(const float* __restrict__ X,
                               const float* __restrict__ Wm,
                               const float* __restrict__ bias,
                               float* __restrict__ Out,
                               int nrows, int relu) {
    const int wave = threadIdx.x >> 5;   // 0..7 -> N tile
    const int lane = threadIdx.x & 31;
    const int half = lane >> 4;          // 0: lanes 0-15, 1: lanes 16-31
    const int l16  = lane & 15;
    const int row0 = blockIdx.x * 16;
    const int n0   = wave * 16;

    // A (16x4 f32): lane l16 holds row M=l16; VGPR0/1 = K = 2*half + {0,1}
    // B (4x16 f32): B[k][n] = Wm[n][k]; lane l16 holds column N=l16, same K striping.
    const float* __restrict__ arow = X  + (size_t)(row0 + l16) * Dfeat + 2 * half;
    const float* __restrict__ brow = Wm + (size_t)(n0   + l16) * Dfeat + 2 * half;

    v8f c = {};
#pragma unroll
    for (int k0 = 0; k0 < Dfeat; k0 += 4) {
        v2f a = *(const v2f*)(arow + k0);
        v2f b = *(const v2f*)(brow + k0);
        // D = A*B + C, f32 x f32 -> f32 accumulate (matches reference precision)
        c = __builtin_amdgcn_wmma_f32_16x16x4_f32(
                /*neg_a=*/false, a, /*neg_b=*/false, b,
                /*c_mod=*/(short)0, c, /*reuse_a=*/false, /*reuse_b=*/false);
    }

    const float bv = bias[n0 + l16];
    // C/D layout: VGPR v, lanes 0-15 -> M=v, lanes 16-31 -> M=v+8; N = n0+l16
    float* __restrict__ orow = Out + (size_t)(row0 + 8 * half) * Dfeat + (n0 + l16);
#pragma unroll
    for (int v = 0; v < 8; ++v) {
        float o = c[v] + bv;
        if (relu) o = fmaxf(o, 0.0f);
        orow[(size_t)v * Dfeat] = o;
    }
}

// 32 lanes per edge, float4 per lane: Out[dst] += H[src] (atomic f32 adds).
__global__ void scatter_add(const float* __restrict__ H,
                            const long long* __restrict__ src,
                            const long long* __restrict__ dst,
                            float* __restrict__ Out, int E) {
    const int t    = blockIdx.x * blockDim.x + threadIdx.x;
    const int edge = t >> 5;
    const int lane = t & 31;
    if (edge >= E) return;
    const long long s = src[edge];
    const long long d = dst[edge];
    const float4 v = *(const float4*)(H + (size_t)s * Dfeat + lane * 4);
    float* o = Out + (size_t)d * Dfeat + lane * 4;
    atomicAdd(o + 0, v.x);
    atomicAdd(o + 1, v.y);
    atomicAdd(o + 2, v.z);
    atomicAdd(o + 3, v.w);
}

__global__ void zero_stats(float* __restrict__ stats) {
    stats[threadIdx.x] = 0.0f;   // launched with 512 threads
}

// blockDim = 128 (one thread per feature column); grid-stride over rows.
__global__ void bn_stats(const float* __restrict__ V, int nrows,
                         float* __restrict__ sum, float* __restrict__ sumsq) {
    const int c = threadIdx.x;
    float s = 0.0f, s2 = 0.0f;
    for (int r = blockIdx.x; r < nrows; r += gridDim.x) {
        const float v = V[(size_t)r * Dfeat + c];
        s  += v;
        s2 += v * v;
    }
    atomicAdd(&sum[c], s);
    atomicAdd(&sumsq[c], s2);
}

__global__ void bn_apply(float* __restrict__ V, int nrows,
                         const float* __restrict__ sum,
                         const float* __restrict__ sumsq,
                         const float* __restrict__ gamma,
                         const float* __restrict__ beta) {
    const size_t total = (size_t)nrows * Dfeat;
    const size_t i = (size_t)blockIdx.x * blockDim.x + threadIdx.x;
    if (i >= total) return;
    const int c = (int)(i & (Dfeat - 1));
    const float n   = (float)nrows;
    const float m   = sum[c] / n;
    const float var = sumsq[c] / n - m * m;   // biased variance (matches jnp.var)
    const float inv = rsqrtf(var + 1e-5f);
    V[i] = (V[i] - m) * inv * gamma[c] + beta[c];
}

extern "C" void kernel_launch(void* const* d_in, const int* in_sizes, int n_in,
                              void* d_out, int out_size, void* d_ws, size_t ws_size,
                              hipStream_t stream) {
    const float*     x_atoms = (const float*)d_in[0];
    const float*     x_bonds = (const float*)d_in[1];
    const long long* ei_ab   = (const long long*)d_in[2];  // [2, E]
    const long long* ei_ba   = (const long long*)d_in[3];  // [2, E]
    const float*     W       = (const float*)d_in[4];
    const float*     b       = (const float*)d_in[5];
    const float*     Wr      = (const float*)d_in[6];
    const float*     br      = (const float*)d_in[7];
    const float*     g_a     = (const float*)d_in[8];
    const float*     bt_a    = (const float*)d_in[9];
    const float*     g_b     = (const float*)d_in[10];
    const float*     bt_b    = (const float*)d_in[11];

    const int NA = 200000, NB = 300000, E = 600000;

    float* out_atoms = (float*)d_out;                         // [NA, 128]
    float* out_bonds = out_atoms + (size_t)NA * Dfeat;        // [NB, 128]

    float* h_atoms = (float*)d_ws;                            // [NA, 128]
    float* h_bonds = h_atoms + (size_t)NA * Dfeat;            // [NB, 128]
    float* stats   = h_bonds + (size_t)NB * Dfeat;            // 512 f32
    // stats layout: [0:128) sum_a, [128:256) sumsq_a, [256:384) sum_b, [384:512) sumsq_b

    // 1) shared lin_rel -> workspace; residual relu path -> d_out (accumulated in place)
    gemm_bias_wmma<<<NA / 16, 256, 0, stream>>>(x_atoms, W,  b,  h_atoms,   NA, 0);
    gemm_bias_wmma<<<NB / 16, 256, 0, stream>>>(x_bonds, W,  b,  h_bonds,   NB, 0);
    gemm_bias_wmma<<<NA / 16, 256, 0, stream>>>(x_atoms, Wr, br, out_atoms, NA, 1);
    gemm_bias_wmma<<<NB / 16, 256, 0, stream>>>(x_bonds, Wr, br, out_bonds, NB, 1);

    zero_stats<<<1, 512, 0, stream>>>(stats);

    // 2) message passing: out_bonds += scatter(h_atoms), out_atoms += scatter(h_bonds)
    const int edgeBlocks = (E * 32 + 255) / 256;
    scatter_add<<<edgeBlocks, 256, 0, stream>>>(h_atoms, ei_ab, ei_ab + E, out_bonds, E);
    scatter_add<<<edgeBlocks, 256, 0, stream>>>(h_bonds, ei_ba, ei_ba + E, out_atoms, E);

    // 3) BatchNorm (training-mode batch stats), in place on d_out
    bn_stats<<<1024, 128, 0, stream>>>(out_atoms, NA, stats,       stats + 128);
    bn_stats<<<1024, 128, 0, stream>>>(out_bonds, NB, stats + 256, stats + 384);

    const size_t totA = (size_t)NA * Dfeat, totB = (size_t)NB * Dfeat;
    bn_apply<<<(unsigned)((totA + 255) / 256), 256, 0, stream>>>(out_atoms, NA, stats,       stats + 128, g_a, bt_a);
    bn_apply<<<(unsigned)((totB + 255) / 256), 256, 0, stream>>>(out_bonds, NB, stats + 256, stats + 384, g_b, bt_b);
}